// MHADecoder_29832842838454
// MI455X (gfx1250) — compile-verified
//
#include <hip/hip_runtime.h>
#include <math.h>

// Problem constants (match the reference)
#define BATCH 32
#define NNODE 512
#define GGRP  512
#define EMBD  256
#define HNUM  16
#define DK    16

typedef __attribute__((ext_vector_type(2))) float v2f;
typedef __attribute__((ext_vector_type(8))) float v8f;

// D = A(16x4, f32) x B(4x16, f32) + C, f32 accumulate.
__device__ __forceinline__ v8f wmma4(v2f a, v2f b, v8f c) {
  return __builtin_amdgcn_wmma_f32_16x16x4_f32(
      /*neg_a=*/false, a, /*neg_b=*/false, b,
      /*c_mod=*/(short)0, c, /*reuse_a=*/false, /*reuse_b=*/false);
}

__device__ __forceinline__ v2f ld2g(const float* __restrict__ p) {
  return *reinterpret_cast<const v2f*>(p);
}
__device__ __forceinline__ v2f ld2s(const float* p) {
  return *reinterpret_cast<const v2f*>(p);
}

// ---------------------------------------------------------------------------
// Kernel 1: per-batch graph mean + q_graph = mean @ Wq_graph^T    [B,256]
// grid = B, block = 256
// ---------------------------------------------------------------------------
__global__ void mean_qg_kernel(const float* __restrict__ EN,
                               const float* __restrict__ WqG,
                               float* __restrict__ qg) {
  __shared__ float sm[EMBD];
  const int b = blockIdx.x;
  const int e = threadIdx.x;
  float s = 0.0f;
  const float* base = EN + (size_t)b * NNODE * EMBD + e;
  for (int n = 0; n < NNODE; ++n) s += base[(size_t)n * EMBD];
  sm[e] = s * (1.0f / (float)NNODE);
  __syncthreads();
  float acc = 0.0f;
  const float* wrow = WqG + (size_t)e * EMBD;
  for (int k = 0; k < EMBD; ++k) acc += sm[k] * wrow[k];
  qg[(size_t)b * EMBD + e] = acc;
}

// ---------------------------------------------------------------------------
// Kernel 2: projections. blockIdx.y: 0 -> K, 1 -> V, 2 -> Q (fused qF+qL+qg)
// One wave computes a 32x64 block (2 M-tiles x 4 N-tiles, 8 accumulators).
// 4 waves/block; 2048 waves per output -> grid.x = 512.
// ---------------------------------------------------------------------------
__global__ void proj_kernel(const float* __restrict__ EN,
                            const float* __restrict__ EF,
                            const float* __restrict__ EL,
                            const float* __restrict__ WqF,
                            const float* __restrict__ WqL,
                            const float* __restrict__ Wk,
                            const float* __restrict__ Wv,
                            const float* __restrict__ qg,
                            float* __restrict__ Q,
                            float* __restrict__ K,
                            float* __restrict__ V) {
  const int widx = threadIdx.x >> 5;
  const int lane = threadIdx.x & 31;
  const int ln   = lane & 15;
  const int kh   = lane >> 4;
  const int wave = blockIdx.x * 4 + widx;   // 0..2047
  const int b    = wave >> 6;               // 64 waves per batch
  const int rem  = wave & 63;
  const int mt   = rem >> 2;                // 0..15 (32-row tiles)
  const int ng   = rem & 3;                 // 0..3  (64-col groups)
  const int row0 = mt * 32 + ln;
  const int sel  = blockIdx.y;

  v8f acc[2][4];
#pragma unroll
  for (int i = 0; i < 2; ++i)
#pragma unroll
    for (int j = 0; j < 4; ++j) acc[i][j] = v8f{0.f,0.f,0.f,0.f,0.f,0.f,0.f,0.f};

  float* Out;
  if (sel < 2) {
    const float* W  = (sel == 0) ? Wk : Wv;
    const float* A0 = EN + (size_t)(b * NNODE + row0) * EMBD;
    const float* A1 = A0 + (size_t)16 * EMBD;
    const float* Br[4];
#pragma unroll
    for (int j = 0; j < 4; ++j) Br[j] = W + (size_t)(ng * 64 + j * 16 + ln) * EMBD;
    for (int k = 0; k < EMBD; k += 4) {
      const v2f a0 = ld2g(A0 + k + 2 * kh);
      const v2f a1 = ld2g(A1 + k + 2 * kh);
#pragma unroll
      for (int j = 0; j < 4; ++j) {
        const v2f bb = ld2g(Br[j] + k + 2 * kh);
        acc[0][j] = wmma4(a0, bb, acc[0][j]);
        acc[1][j] = wmma4(a1, bb, acc[1][j]);
      }
    }
    Out = (sel == 0) ? K : V;
  } else {
    const float* A0 = EF + (size_t)(b * GGRP + row0) * EMBD;
    const float* A1 = A0 + (size_t)16 * EMBD;
    const float* Br[4];
#pragma unroll
    for (int j = 0; j < 4; ++j) Br[j] = WqF + (size_t)(ng * 64 + j * 16 + ln) * EMBD;
    for (int k = 0; k < EMBD; k += 4) {
      const v2f a0 = ld2g(A0 + k + 2 * kh);
      const v2f a1 = ld2g(A1 + k + 2 * kh);
#pragma unroll
      for (int j = 0; j < 4; ++j) {
        const v2f bb = ld2g(Br[j] + k + 2 * kh);
        acc[0][j] = wmma4(a0, bb, acc[0][j]);
        acc[1][j] = wmma4(a1, bb, acc[1][j]);
      }
    }
    const float* C0 = EL + (size_t)(b * GGRP + row0) * EMBD;
    const float* C1 = C0 + (size_t)16 * EMBD;
#pragma unroll
    for (int j = 0; j < 4; ++j) Br[j] = WqL + (size_t)(ng * 64 + j * 16 + ln) * EMBD;
    for (int k = 0; k < EMBD; k += 4) {
      const v2f a0 = ld2g(C0 + k + 2 * kh);
      const v2f a1 = ld2g(C1 + k + 2 * kh);
#pragma unroll
      for (int j = 0; j < 4; ++j) {
        const v2f bb = ld2g(Br[j] + k + 2 * kh);
        acc[0][j] = wmma4(a0, bb, acc[0][j]);
        acc[1][j] = wmma4(a1, bb, acc[1][j]);
      }
    }
#pragma unroll
    for (int j = 0; j < 4; ++j) {
      const float qv = qg[(size_t)b * EMBD + ng * 64 + j * 16 + ln];
#pragma unroll
      for (int i = 0; i < 2; ++i)
#pragma unroll
        for (int r = 0; r < 8; ++r) acc[i][j][r] += qv;
    }
    Out = Q;
  }

#pragma unroll
  for (int i = 0; i < 2; ++i)
#pragma unroll
    for (int j = 0; j < 4; ++j) {
      const int col = ng * 64 + j * 16 + ln;
#pragma unroll
      for (int r = 0; r < 8; ++r) {
        const int row = mt * 32 + i * 16 + r + 8 * kh;
        Out[(size_t)(b * NNODE + row) * EMBD + col] = acc[i][j][r];
      }
    }
}

// ---------------------------------------------------------------------------
// Kernel 3: attention per (b, h, g-tile). block = 256 (8 waves).
// scores (16x512) -> softmax -> w @ V_h -> out_concat slice.
// ---------------------------------------------------------------------------
__global__ void attn_kernel(const float* __restrict__ Q,
                            const float* __restrict__ K,
                            const float* __restrict__ V,
                            const float* __restrict__ MASK,
                            float* __restrict__ outc) {
  __shared__ float sc[16 * 512];      // scores / weights
  __shared__ float part[8 * 256];     // per-wave partial 16x16 outputs
  __shared__ float redA[16][16];
  __shared__ float redB[16][16];

  const int gt = blockIdx.x, h = blockIdx.y, b = blockIdx.z;
  const int t    = threadIdx.x;
  const int widx = t >> 5;
  const int lane = t & 31;
  const int ln   = lane & 15;
  const int kh   = lane >> 4;

  // ---- scores = (q . k) / 4 + mask : A fragment shared across 4 n-tiles ----
  {
    const float* Arow = Q + (size_t)(b * GGRP + gt * 16 + ln) * EMBD + h * DK;
    const float* Br[4];
#pragma unroll
    for (int j = 0; j < 4; ++j)
      Br[j] = K + (size_t)(b * NNODE + (widx * 4 + j) * 16 + ln) * EMBD + h * DK;
    v8f acc[4];
#pragma unroll
    for (int j = 0; j < 4; ++j) acc[j] = v8f{0.f,0.f,0.f,0.f,0.f,0.f,0.f,0.f};
#pragma unroll
    for (int k = 0; k < DK; k += 4) {
      const v2f a = ld2g(Arow + k + 2 * kh);
#pragma unroll
      for (int j = 0; j < 4; ++j)
        acc[j] = wmma4(a, ld2g(Br[j] + k + 2 * kh), acc[j]);
    }
#pragma unroll
    for (int j = 0; j < 4; ++j) {
      const int nb = (widx * 4 + j) * 16;
#pragma unroll
      for (int r = 0; r < 8; ++r) {
        const int row = r + 8 * kh;
        const int g   = gt * 16 + row;
        sc[row * 512 + nb + ln] =
            acc[j][r] * 0.25f + MASK[(size_t)(b * GGRP + g) * NNODE + nb + ln];
      }
    }
  }
  __syncthreads();

  // ---- row softmax over N = 512 ----
  const int row = t >> 4, l16 = t & 15;
  float* srow = sc + row * 512;
  float mx = -INFINITY;
  for (int j = l16; j < 512; j += 16) mx = fmaxf(mx, srow[j]);
  redA[row][l16] = mx;
  __syncthreads();
  if (l16 == 0) {
    float m2 = redA[row][0];
    for (int i = 1; i < 16; ++i) m2 = fmaxf(m2, redA[row][i]);
    redA[row][0] = m2;
  }
  __syncthreads();
  const float rmax = redA[row][0];
  float ssum = 0.0f;
  for (int j = l16; j < 512; j += 16) {
    const float e = __expf(srow[j] - rmax);
    srow[j] = e;
    ssum += e;
  }
  redB[row][l16] = ssum;
  __syncthreads();
  if (l16 == 0) {
    float s2 = 0.0f;
    for (int i = 0; i < 16; ++i) s2 += redB[row][i];
    redB[row][0] = s2;
  }
  __syncthreads();
  const float inv = 1.0f / redB[row][0];
  for (int j = l16; j < 512; j += 16) srow[j] *= inv;
  __syncthreads();

  // ---- out = w @ V_h, per-wave partials over 64 columns each ----
  v8f acc = {0.f,0.f,0.f,0.f,0.f,0.f,0.f,0.f};
#pragma unroll
  for (int nti = 0; nti < 4; ++nti) {
    const int nb = (widx * 4 + nti) * 16;
#pragma unroll
    for (int k = 0; k < 16; k += 4) {
      const int kk = nb + k + 2 * kh;           // global n index of this k-pair
      v2f a = ld2s(sc + ln * 512 + kk);         // w[row=ln][kk], w[ln][kk+1]
      const float* vp = V + (size_t)(b * NNODE + kk) * EMBD + h * DK + ln;
      v2f bb;
      bb.x = vp[0];
      bb.y = vp[EMBD];
      acc = wmma4(a, bb, acc);
    }
  }
#pragma unroll
  for (int r = 0; r < 8; ++r)
    part[widx * 256 + (r + 8 * kh) * 16 + ln] = acc[r];
  __syncthreads();

  {
    const int orow = t >> 4, ocol = t & 15;
    float s = 0.0f;
#pragma unroll
    for (int w = 0; w < 8; ++w) s += part[w * 256 + orow * 16 + ocol];
    outc[(size_t)(b * GGRP + gt * 16 + orow) * EMBD + h * DK + ocol] = s;
  }
}

// ---------------------------------------------------------------------------
// Kernel 4: mh = out_concat @ Wc^T + bc. Same 32x64 blocking as proj_kernel.
// ---------------------------------------------------------------------------
__global__ void mh_kernel(const float* __restrict__ outc,
                          const float* __restrict__ Wc,
                          const float* __restrict__ bc,
                          float* __restrict__ mh) {
  const int widx = threadIdx.x >> 5;
  const int lane = threadIdx.x & 31;
  const int ln   = lane & 15;
  const int kh   = lane >> 4;
  const int wave = blockIdx.x * 4 + widx;   // 0..2047
  const int b    = wave >> 6;
  const int rem  = wave & 63;
  const int mt   = rem >> 2;
  const int ng   = rem & 3;
  const int row0 = mt * 32 + ln;

  v8f acc[2][4];
#pragma unroll
  for (int i = 0; i < 2; ++i)
#pragma unroll
    for (int j = 0; j < 4; ++j) acc[i][j] = v8f{0.f,0.f,0.f,0.f,0.f,0.f,0.f,0.f};

  const float* A0 = outc + (size_t)(b * GGRP + row0) * EMBD;
  const float* A1 = A0 + (size_t)16 * EMBD;
  const float* Br[4];
#pragma unroll
  for (int j = 0; j < 4; ++j) Br[j] = Wc + (size_t)(ng * 64 + j * 16 + ln) * EMBD;
  for (int k = 0; k < EMBD; k += 4) {
    const v2f a0 = ld2g(A0 + k + 2 * kh);
    const v2f a1 = ld2g(A1 + k + 2 * kh);
#pragma unroll
    for (int j = 0; j < 4; ++j) {
      const v2f bb = ld2g(Br[j] + k + 2 * kh);
      acc[0][j] = wmma4(a0, bb, acc[0][j]);
      acc[1][j] = wmma4(a1, bb, acc[1][j]);
    }
  }
#pragma unroll
  for (int i = 0; i < 2; ++i)
#pragma unroll
    for (int j = 0; j < 4; ++j) {
      const int col  = ng * 64 + j * 16 + ln;
      const float bv = bc[col];
#pragma unroll
      for (int r = 0; r < 8; ++r) {
        const int row = mt * 32 + i * 16 + r + 8 * kh;
        mh[(size_t)(b * GGRP + row) * EMBD + col] = acc[i][j][r] + bv;
      }
    }
}

// ---------------------------------------------------------------------------
// Kernel 5: score2 = mh @ EN^T / 16 -> clip*tanh + mask -> row softmax -> out
// One block per (g-tile, b). block = 256 (8 waves, 4 n-tiles each, shared A).
// ---------------------------------------------------------------------------
__global__ void final_kernel(const float* __restrict__ mh,
                             const float* __restrict__ EN,
                             const float* __restrict__ MASK,
                             float* __restrict__ out) {
  __shared__ float sc[16 * 512];
  __shared__ float redA[16][16];
  __shared__ float redB[16][16];

  const int gt = blockIdx.x, b = blockIdx.y;
  const int t    = threadIdx.x;
  const int widx = t >> 5;
  const int lane = t & 31;
  const int ln   = lane & 15;
  const int kh   = lane >> 4;

  {
    const float* Arow = mh + (size_t)(b * GGRP + gt * 16 + ln) * EMBD;
    const float* Br[4];
#pragma unroll
    for (int j = 0; j < 4; ++j)
      Br[j] = EN + (size_t)(b * NNODE + (widx * 4 + j) * 16 + ln) * EMBD;
    v8f acc[4];
#pragma unroll
    for (int j = 0; j < 4; ++j) acc[j] = v8f{0.f,0.f,0.f,0.f,0.f,0.f,0.f,0.f};
    for (int k = 0; k < EMBD; k += 4) {
      const v2f a = ld2g(Arow + k + 2 * kh);
#pragma unroll
      for (int j = 0; j < 4; ++j)
        acc[j] = wmma4(a, ld2g(Br[j] + k + 2 * kh), acc[j]);
    }
#pragma unroll
    for (int j = 0; j < 4; ++j) {
      const int nb = (widx * 4 + j) * 16;
#pragma unroll
      for (int r = 0; r < 8; ++r) {
        const int row = r + 8 * kh;
        const int g   = gt * 16 + row;
        const float s2 = acc[j][r] * (1.0f / 16.0f);  // 1/sqrt(EMB)
        sc[row * 512 + nb + ln] =
            10.0f * tanhf(s2) + MASK[(size_t)(b * GGRP + g) * NNODE + nb + ln];
      }
    }
  }
  __syncthreads();

  const int row = t >> 4, l16 = t & 15;
  float* srow = sc + row * 512;
  float mx = -INFINITY;
  for (int j = l16; j < 512; j += 16) mx = fmaxf(mx, srow[j]);
  redA[row][l16] = mx;
  __syncthreads();
  if (l16 == 0) {
    float m2 = redA[row][0];
    for (int i = 1; i < 16; ++i) m2 = fmaxf(m2, redA[row][i]);
    redA[row][0] = m2;
  }
  __syncthreads();
  const float rmax = redA[row][0];
  float ssum = 0.0f;
  for (int j = l16; j < 512; j += 16) {
    const float e = __expf(srow[j] - rmax);
    srow[j] = e;
    ssum += e;
  }
  redB[row][l16] = ssum;
  __syncthreads();
  if (l16 == 0) {
    float s2 = 0.0f;
    for (int i = 0; i < 16; ++i) s2 += redB[row][i];
    redB[row][0] = s2;
  }
  __syncthreads();
  const float inv = 1.0f / redB[row][0];
  float* orow = out + (size_t)(b * GGRP + gt * 16 + row) * NNODE;
  for (int j = l16; j < 512; j += 16) orow[j] = srow[j] * inv;
}

// ---------------------------------------------------------------------------
extern "C" void kernel_launch(void* const* d_in, const int* in_sizes, int n_in,
                              void* d_out, int out_size, void* d_ws, size_t ws_size,
                              hipStream_t stream) {
  const float* EN   = (const float*)d_in[0];
  const float* EF   = (const float*)d_in[1];
  const float* EL   = (const float*)d_in[2];
  const float* MASK = (const float*)d_in[3];
  const float* WqG  = (const float*)d_in[4];
  const float* WqF  = (const float*)d_in[5];
  const float* WqL  = (const float*)d_in[6];
  const float* Wk   = (const float*)d_in[7];
  const float* Wv   = (const float*)d_in[8];
  const float* Wc   = (const float*)d_in[9];
  const float* bc   = (const float*)d_in[10];
  float* out = (float*)d_out;

  // Workspace layout (floats)
  float* ws   = (float*)d_ws;
  float* qg   = ws;                               // 32*256
  float* Qb   = qg   + (size_t)BATCH * EMBD;      // 32*512*256 each below
  float* Kb   = Qb   + (size_t)BATCH * GGRP * EMBD;
  float* Vb   = Kb   + (size_t)BATCH * NNODE * EMBD;
  float* outc = Vb   + (size_t)BATCH * NNODE * EMBD;
  float* mhb  = outc + (size_t)BATCH * GGRP * EMBD;

  mean_qg_kernel<<<BATCH, 256, 0, stream>>>(EN, WqG, qg);

  dim3 gp(512, 3, 1);  // 2048 waves per output, 4 waves/block
  proj_kernel<<<gp, 128, 0, stream>>>(EN, EF, EL, WqF, WqL, Wk, Wv, qg, Qb, Kb, Vb);

  dim3 ga(GGRP / 16, HNUM, BATCH);
  attn_kernel<<<ga, 256, 0, stream>>>(Qb, Kb, Vb, MASK, outc);

  mh_kernel<<<512, 128, 0, stream>>>(outc, Wc, bc, mhb);

  dim3 gf(GGRP / 16, BATCH, 1);
  final_kernel<<<gf, 256, 0, stream>>>(mhb, EN, MASK, out);
}